// P_Hologen_70317204570282
// MI455X (gfx1250) — compile-verified
//
#include <hip/hip_runtime.h>
#include <math.h>

typedef __attribute__((ext_vector_type(2))) float v2f;
typedef __attribute__((ext_vector_type(8))) float v8f;

#define PI_CLIP 3.14159265358979f

// ---------------------------------------------------------------------------
// poh = clip(poh_dec, -pi, pi)
// ---------------------------------------------------------------------------
__global__ void k_clip(const float* __restrict__ in, float* __restrict__ out, int n) {
    int t = blockIdx.x * blockDim.x + threadIdx.x;
    if (t < n) {
        float x = in[t];
        out[t] = fminf(fmaxf(x, -PI_CLIP), PI_CLIP);
    }
}

// ---------------------------------------------------------------------------
// ssq[j] = sum_k emb[j][k]^2   (512 codebook entries, D=256)
// ---------------------------------------------------------------------------
__global__ void k_sumsq(const float* __restrict__ emb, float* __restrict__ ssq) {
    int j = blockIdx.x * blockDim.x + threadIdx.x;
    if (j < 512) {
        const float* e = emb + (size_t)j * 256;
        float s = 0.f;
        for (int k = 0; k < 256; ++k) { float v = e[k]; s = fmaf(v, v, s); }
        ssq[j] = s;
    }
}

// ---------------------------------------------------------------------------
// VQ argmin via V_WMMA_F32_16X16X4_F32.
// One wave handles 16 rows of ze (65536x256) against all 512 codes.
// score(i,j) = ssq[j] - 2 * (ze_i . emb_j)   (||ze||^2 constant per row).
//
// A-tile (16 rows x 256 k) is staged once per wave into LDS, packed so the
// WMMA A fragment (lane m = row, khi = K-pair select) is one ds_load_b64:
//   element (k,row) stored at float index (k>>1)*32 + row*2 + (k&1).
// B comes straight from global (emb is 512KB -> L2 resident).
// Argmin reduction is fully branchless (v_cndmask, no EXEC branching).
// ---------------------------------------------------------------------------
__global__ void __launch_bounds__(128) k_vq(const float* __restrict__ zex,
                                            const float* __restrict__ emb,
                                            const float* __restrict__ ssq,
                                            int* __restrict__ idx) {
    __shared__ float As_all[4 * 4096];   // 64 KB: 4 waves x (16 rows x 256 k)

    const int lane = threadIdx.x & 31;
    const int wave = threadIdx.x >> 5;
    const int rowBlock = blockIdx.x * 4 + wave;   // 0..4095
    const int i0 = rowBlock << 4;
    const int m   = lane & 15;
    const int khi = lane >> 4;

    // z_e_x layout: [b][d][h][w] -> b*2^20 + k*4096 + p ; rows are contiguous p
    const int b  = i0 >> 12;
    const int p0 = i0 & 4095;
    const float* zeBase = zex + (size_t)b * 1048576 + p0;   // + k*4096 + row

    float* As = As_all + wave * 4096;

    // ---- stage A tile: coalesced global_load_b128 (4 rows per lane, 8 k/iter)
    {
        const int k_off = lane >> 2;        // 0..7
        const int r0    = (lane & 3) * 4;   // 0,4,8,12
        for (int kk = 0; kk < 256; kk += 8) {
            const int k = kk + k_off;
            const float4 v = *(const float4*)(zeBase + (size_t)k * 4096 + r0);
            float* dst = &As[(k >> 1) * 32 + (k & 1)];
            dst[(r0 + 0) * 2] = v.x;
            dst[(r0 + 1) * 2] = v.y;
            dst[(r0 + 2) * 2] = v.z;
            dst[(r0 + 3) * 2] = v.w;
        }
    }
    __syncthreads();

    float bestS[8];
    int   bestJ[8];
#pragma unroll
    for (int v = 0; v < 8; ++v) { bestS[v] = 3.4e38f; bestJ[v] = 0; }

    const float2* As2 = (const float2*)As;   // fragment (k0,m) at (k0>>1)*16 + m

    for (int jt = 0; jt < 32; ++jt) {
        const int jbase = jt << 4;
        const float* embrow = emb + (size_t)(jbase + m) * 256;  // B: n = m
        const float  sj     = ssq[jbase + m];

        // prefetch next codebook tile (global_prefetch_b8)
        const int pj = (jbase + 16 < 512) ? (jbase + 16) : 496;
        __builtin_prefetch(emb + (size_t)(pj + m) * 256, 0, 1);

        v8f acc = {0.f, 0.f, 0.f, 0.f, 0.f, 0.f, 0.f, 0.f};
#pragma unroll 8
        for (int kb = 0; kb < 256; kb += 4) {
            const int k0 = kb + (khi << 1);
            const float2 af = As2[(k0 >> 1) * 16 + m];          // ds_load_b64
            const float2 bf = *(const float2*)(embrow + k0);    // global_load_b64
            v2f a, bb;
            a.x = af.x;  a.y = af.y;
            bb.x = bf.x; bb.y = bf.y;
            acc = __builtin_amdgcn_wmma_f32_16x16x4_f32(
                /*neg_a=*/false, a, /*neg_b=*/false, bb,
                /*c_mod=*/(short)0, acc, /*reuse_a=*/false, /*reuse_b=*/false);
        }

        // branchless per-row argmin across the 16 lanes of each half-wave
#pragma unroll
        for (int v = 0; v < 8; ++v) {
            float sc = fmaf(-2.f, acc[v], sj);
            int   jj = jbase + m;
#pragma unroll
            for (int d = 8; d; d >>= 1) {   // masks 8,4,2,1 keep halves separate
                const float os = __shfl_xor(sc, d, 32);
                const int   oj = __shfl_xor(jj, d, 32);
                const int take = (int)(os < sc) | ((int)(os == sc) & (int)(oj < jj));
                sc = take ? os : sc;
                jj = take ? oj : jj;
            }
            const int take = (int)(sc < bestS[v]) |
                             ((int)(sc == bestS[v]) & (int)(jj < bestJ[v]));
            bestS[v] = take ? sc : bestS[v];
            bestJ[v] = take ? jj : bestJ[v];
        }
    }
    if (m == 0) {   // lane 0 -> rows 0..7, lane 16 -> rows 8..15
#pragma unroll
        for (int v = 0; v < 8; ++v) idx[i0 + (khi << 3) + v] = bestJ[v];
    }
}

// ---------------------------------------------------------------------------
// z_q_x[b][d][h][w] = emb[idx[b,h,w]][d]
// ---------------------------------------------------------------------------
__global__ void k_gather(const int* __restrict__ idx, const float* __restrict__ emb,
                         float* __restrict__ zqx, int n) {
    int t = blockIdx.x * blockDim.x + threadIdx.x;
    if (t >= n) return;
    int p = t & 4095;
    int d = (t >> 12) & 255;
    int b = t >> 20;
    int j = idx[(b << 12) + p];
    zqx[t] = emb[(size_t)j * 256 + d];
}

// ---------------------------------------------------------------------------
// 512-point radix-2 in-LDS FFT; 256 threads, one butterfly per thread/stage.
// sign = -1 forward, +1 inverse (unnormalized).
// ---------------------------------------------------------------------------
__device__ __forceinline__ void fft512(float2* s, int tid, float sign) {
#pragma unroll
    for (int h = 0; h < 2; ++h) {           // 9-bit bit-reversal permutation
        int i = tid + h * 256;
        int j = (int)(__brev((unsigned)i) >> 23);
        if (i < j) { float2 t = s[i]; s[i] = s[j]; s[j] = t; }
    }
    __syncthreads();
#pragma unroll
    for (int len = 2; len <= 512; len <<= 1) {
        const int half = len >> 1;
        const int grp  = tid / half;
        const int pos  = tid - grp * half;
        const int ia   = grp * len + pos;
        const int ib   = ia + half;
        float sn, cs;
        __sincosf(sign * 6.28318530717958648f * (float)pos / (float)len, &sn, &cs);
        const float2 u = s[ia];
        const float2 w = s[ib];
        const float tr = cs * w.x - sn * w.y;
        const float ti = cs * w.y + sn * w.x;
        s[ia] = make_float2(u.x + tr, u.y + ti);
        s[ib] = make_float2(u.x - tr, u.y - ti);
        __syncthreads();
    }
}

// Pass 1: build padded field e^{i clip(poh)} and forward-FFT each row.
__global__ void __launch_bounds__(256) k_fft_rows_fwd(const float* __restrict__ poh_dec,
                                                      float2* __restrict__ A) {
    __shared__ float2 s[512];
    const int r = blockIdx.x;     // 0..511
    const int b = blockIdx.y;     // 0..15
    const int tid = threadIdx.x;
    float2* Arow = A + ((size_t)b * 512 + r) * 512;
    if (r < 128 || r >= 384) {    // zero-padded rows: FFT is zero
        const float2 z = make_float2(0.f, 0.f);
        Arow[tid] = z; Arow[tid + 256] = z;
        return;
    }
    const float* prow = poh_dec + ((size_t)b * 256 + (r - 128)) * 256;
#pragma unroll
    for (int h = 0; h < 2; ++h) {
        int c = tid + h * 256;
        float2 v = make_float2(0.f, 0.f);
        if (c >= 128 && c < 384) {
            float phi = prow[c - 128];
            phi = fminf(fmaxf(phi, -PI_CLIP), PI_CLIP);
            __sincosf(phi, &v.y, &v.x);   // v = (cos, sin)
        }
        s[c] = v;
    }
    __syncthreads();
    fft512(s, tid, -1.0f);
    Arow[tid]       = s[tid];
    Arow[tid + 256] = s[tid + 256];
}

// Pass 2: per column: forward FFT, multiply by H[r][c], inverse FFT (in place).
__global__ void __launch_bounds__(256) k_fft_cols(float2* __restrict__ A,
                                                  const float2* __restrict__ H) {
    __shared__ float2 s[512];
    const int c = blockIdx.x;     // 0..511
    const int b = blockIdx.y;
    const int tid = threadIdx.x;
    float2* Abase = A + (size_t)b * 512 * 512 + c;
#pragma unroll
    for (int h = 0; h < 2; ++h) {
        int r = tid + h * 256;
        s[r] = Abase[(size_t)r * 512];
    }
    __syncthreads();
    fft512(s, tid, -1.0f);
#pragma unroll
    for (int h = 0; h < 2; ++h) {
        int r = tid + h * 256;
        float2 u  = s[r];
        float2 hh = H[(size_t)r * 512 + c];
        s[r] = make_float2(u.x * hh.x - u.y * hh.y, u.x * hh.y + u.y * hh.x);
    }
    __syncthreads();
    fft512(s, tid, +1.0f);
#pragma unroll
    for (int h = 0; h < 2; ++h) {
        int r = tid + h * 256;
        Abase[(size_t)r * 512] = s[r];
    }
}

// Pass 3: inverse row FFT of the 256 center rows, crop cols, |u| * sqrt(0.95),
// combined ortho scale 1/512^2.
__global__ void __launch_bounds__(256) k_fft_rows_inv_crop(const float2* __restrict__ A,
                                                           float* __restrict__ recon) {
    __shared__ float2 s[512];
    const int rr  = blockIdx.x;   // 0..255
    const int b   = blockIdx.y;
    const int tid = threadIdx.x;
    const int r   = rr + 128;
    const float2* Arow = A + ((size_t)b * 512 + r) * 512;
    s[tid]       = Arow[tid];
    s[tid + 256] = Arow[tid + 256];
    __syncthreads();
    fft512(s, tid, +1.0f);
    const float sc = 0.974679434480896f / 262144.0f;   // sqrt(0.95) / 512^2
    float* orow = recon + ((size_t)b * 256 + rr) * 256;
#pragma unroll
    for (int h = 0; h < 2; ++h) {
        int c = tid + h * 256;
        if (c >= 128 && c < 384) {
            float2 u = s[c];
            orow[c - 128] = sqrtf(u.x * u.x + u.y * u.y) * sc;
        }
    }
}

// ---------------------------------------------------------------------------
extern "C" void kernel_launch(void* const* d_in, const int* in_sizes, int n_in,
                              void* d_out, int out_size, void* d_ws, size_t ws_size,
                              hipStream_t stream) {
    (void)in_sizes; (void)n_in; (void)out_size; (void)d_ws; (void)ws_size;

    const float*  zex     = (const float*)d_in[0];   // (16,256,64,64)
    const float*  poh_dec = (const float*)d_in[1];   // (16,1,256,256)
    const float*  emb     = (const float*)d_in[2];   // (512,256)
    const float2* H       = (const float2*)d_in[3];  // (512,512) complex64

    float* out       = (float*)d_out;
    float* out_poh   = out;                 //  1,048,576
    float* out_recon = out + 1048576;       //  1,048,576
    float* out_zex   = out + 2097152;       // 16,777,216
    float* out_zqx   = out + 18874368;      // 16,777,216

    // Scratch aliased into output regions that are (re)written later:
    float2* A      = (float2*)out_zqx;          // 32 MB FFT intermediate
    float*  ssq    = out_zex;                   // 512 floats
    int*    idxbuf = (int*)(out_zex + 512);     // 65536 ints

    // 1) poh output
    k_clip<<<1048576 / 256, 256, 0, stream>>>(poh_dec, out_poh, 1048576);

    // 2) hologram reconstruction (scratch in out_zqx region)
    k_fft_rows_fwd<<<dim3(512, 16), 256, 0, stream>>>(poh_dec, A);
    k_fft_cols<<<dim3(512, 16), 256, 0, stream>>>(A, H);
    k_fft_rows_inv_crop<<<dim3(256, 16), 256, 0, stream>>>(A, out_recon);

    // 3) VQ (scratch in out_zex region)
    k_sumsq<<<2, 256, 0, stream>>>(emb, ssq);
    k_vq<<<1024, 128, 0, stream>>>(zex, emb, ssq, idxbuf);

    // 4) z_q_x gather (overwrites FFT scratch — ordered after pass 3)
    k_gather<<<(16 * 1048576) / 256, 256, 0, stream>>>(idxbuf, emb, out_zqx, 16 * 1048576);

    // 5) z_e_x passthrough copy (overwrites ssq/idx scratch — ordered last)
    hipMemcpyAsync(out_zex, zex, (size_t)16 * 1048576 * sizeof(float),
                   hipMemcpyDeviceToDevice, stream);
}